// Partitioner_11527692223039
// MI455X (gfx1250) — compile-verified
//
#include <hip/hip_runtime.h>
#include <hip/hip_bf16.h>

typedef float v2f __attribute__((ext_vector_type(2)));
typedef float v8f __attribute__((ext_vector_type(8)));

#define KBOX 1024
#define EPSF 1e-4f
#define HPOSED_OUT (KBOX * 8 * 3)   // 24576 floats, first output

// ---- workspace layout (bytes) ----
#define WS_HPOSED 0          // [K][8][4] f32   131072 B
#define WS_INVT   131072     // [K][4][4] f32    65536 B
#define WS_RAW    196608     // [K][32]  u32    131072 B
#define WS_FIN    327680     // [K][32]  u32    131072 B
#define WS_CNT    458752     // [K] i32           4096 B
#define WS_OFF    462848     // [K] i32           4096 B

static __device__ __forceinline__ unsigned ballot32(bool b) {
    return __builtin_amdgcn_ballot_w32(b);
}

// Per-D-tile inside test. D layout: VGPR r holds (M=r, N=lane) for lanes 0-15
// and (M=r+8, N=lane-16) for lanes 16-31; N = 4*kk + i (box kk, coordinate i).
// Ballot each compare separately -> v_cmp straight into SGPRs, AND in SALU.
static __device__ __forceinline__ unsigned inside_bits(v8f d, float lo, float hi) {
    unsigned acc = 0;
#pragma unroll
    for (int r = 0; r < 8; ++r) {
        unsigned bgt  = ballot32(d[r] > lo);
        unsigned blt  = ballot32(d[r] < hi);
        unsigned ball = bgt & blt;
        unsigned and3 = ball & (ball >> 1) & (ball >> 2);
        acc |= and3 & 0x11111111u;   // bits 4*kk (box l0) and 16+4*kk (box l0+1)
    }
    return acc;
}

// ---------------------------------------------------------------------------
// Kernel 1: per-box prep. Pose the 8 corners (hposed to ws + xyz to d_out) and
// build the affine inverse transform (exact: bottom row is 0,0,0,1).
// ---------------------------------------------------------------------------
__global__ void prep_kernel(const float* __restrict__ bb_min,
                            const float* __restrict__ bb_max,
                            const float* __restrict__ xf,
                            float* __restrict__ hposed,
                            float* __restrict__ invT,
                            float* __restrict__ out_hposed) {
    int k = blockIdx.x * blockDim.x + threadIdx.x;
    if (k >= KBOX) return;

    float T[16];
#pragma unroll
    for (int i = 0; i < 16; ++i) T[i] = xf[k * 16 + i];

    float mnx = bb_min[k * 3 + 0], mny = bb_min[k * 3 + 1], mnz = bb_min[k * 3 + 2];
    float mxx = bb_max[k * 3 + 0], mxy = bb_max[k * 3 + 1], mxz = bb_max[k * 3 + 2];

#pragma unroll
    for (int c = 0; c < 8; ++c) {
        // _BITS[c] = [(c>>2)&1, (c>>1)&1, c&1]
        float cx = ((c >> 2) & 1) ? mxx : mnx;
        float cy = ((c >> 1) & 1) ? mxy : mny;
        float cz = ((c >> 0) & 1) ? mxz : mnz;
#pragma unroll
        for (int i = 0; i < 4; ++i) {
            float v = T[i * 4 + 0] * cx + T[i * 4 + 1] * cy + T[i * 4 + 2] * cz + T[i * 4 + 3];
            hposed[(k * 8 + c) * 4 + i] = v;
            if (i < 3) out_hposed[(k * 8 + c) * 3 + i] = v;
        }
    }

    float r00 = T[0], r01 = T[1], r02 = T[2],  t0 = T[3];
    float r10 = T[4], r11 = T[5], r12 = T[6],  t1 = T[7];
    float r20 = T[8], r21 = T[9], r22 = T[10], t2 = T[11];
    float det = r00 * (r11 * r22 - r12 * r21)
              - r01 * (r10 * r22 - r12 * r20)
              + r02 * (r10 * r21 - r11 * r20);
    float id = 1.0f / det;
    float i00 = (r11 * r22 - r12 * r21) * id;
    float i01 = (r02 * r21 - r01 * r22) * id;
    float i02 = (r01 * r12 - r02 * r11) * id;
    float i10 = (r12 * r20 - r10 * r22) * id;
    float i11 = (r00 * r22 - r02 * r20) * id;
    float i12 = (r02 * r10 - r00 * r12) * id;
    float i20 = (r10 * r21 - r11 * r20) * id;
    float i21 = (r01 * r20 - r00 * r21) * id;
    float i22 = (r00 * r11 - r01 * r10) * id;
    float ti0 = -(i00 * t0 + i01 * t1 + i02 * t2);
    float ti1 = -(i10 * t0 + i11 * t1 + i12 * t2);
    float ti2 = -(i20 * t0 + i21 * t1 + i22 * t2);

    float* iv = invT + k * 16;
    iv[0] = i00; iv[1]  = i01; iv[2]  = i02; iv[3]  = ti0;
    iv[4] = i10; iv[5]  = i11; iv[6]  = i12; iv[7]  = ti1;
    iv[8] = i20; iv[9]  = i21; iv[10] = i22; iv[11] = ti2;
    iv[12] = 0.f; iv[13] = 0.f; iv[14] = 0.f; iv[15] = 1.f;
}

// ---------------------------------------------------------------------------
// Kernel 2: O(K^2) pair test via V_WMMA_F32_16X16X4_F32.
// Block = 4 "k" boxes (B operand + bounds are loop-invariant).
// Wave  = 128 "l" boxes, 2 per WMMA, unrolled x4 and software-pipelined so
// each s_wait covers a full iteration of 4 WMMAs + ballot reduction.
// ---------------------------------------------------------------------------
__global__ void __launch_bounds__(256)
pairs_wmma_kernel(const float* __restrict__ hposed,
                  const float* __restrict__ invT,
                  const float* __restrict__ bb_min,
                  const float* __restrict__ bb_max,
                  unsigned* __restrict__ raw) {
    const int k0   = blockIdx.x * 4;        // 256 blocks cover 1024 k rows
    const int wave = threadIdx.x >> 5;      // 0..7
    const int lane = threadIdx.x & 31;
    const int n     = lane & 15;
    const int khalf = lane >> 4;            // 0: K-dims {0,1}, 1: K-dims {2,3}

    // B operand: B[j][4*kk+i] = invT[k0+kk][i][j]
    const int kk = n >> 2;
    const int ci = n & 3;
    const float* brow = invT + ((k0 + kk) * 4 + ci) * 4;
    v2f b;
    b.x = brow[2 * khalf + 0];
    b.y = brow[2 * khalf + 1];

    // per-lane interval bounds for (box k0+kk, coordinate ci)
    float lo, hi;
    if (ci < 3) {
        lo = bb_min[(k0 + kk) * 3 + ci] + EPSF;
        hi = bb_max[(k0 + kk) * 3 + ci] - EPSF;
    } else {                                 // homogeneous w lane: never gates
        lo = -3.0e38f;
        hi = 3.0e38f;
    }

    const int m = lane & 15;                 // A-matrix row for this lane
    const int lbase = wave * 128;
    const int slot = (lane & 3) * 4;         // safe shifts on every lane
    unsigned rowword = 0;

    // A row pointer for this lane: hposed + (l0*8 + m)*4 + 2*khalf.
    // l0 advances by 2 boxes per WMMA -> +64 floats; by 8 per iteration -> +256.
    const float* abase = hposed + (lbase * 8 + m) * 4 + 2 * khalf;

    // preload iteration 0
    v2f a0 = *(const v2f*)(abase + 0);
    v2f a1 = *(const v2f*)(abase + 64);
    v2f a2 = *(const v2f*)(abase + 128);
    v2f a3 = *(const v2f*)(abase + 192);

    for (int it = 0; it < 16; ++it) {
        const int l0 = lbase + it * 8;
        // issue next iteration's loads first (it==15 reloads it=0: harmless,
        // keeps control flow uniform for the WMMAs)
        const float* anext = abase + ((it + 1) & 15) * 256;
        v2f n0 = *(const v2f*)(anext + 0);
        v2f n1 = *(const v2f*)(anext + 64);
        v2f n2 = *(const v2f*)(anext + 128);
        v2f n3 = *(const v2f*)(anext + 192);
        __builtin_prefetch(abase + ((it + 2) & 15) * 256, 0, 3);

        v8f z = {};
        v8f d0 = __builtin_amdgcn_wmma_f32_16x16x4_f32(false, a0, false, b, (short)0, z, false, false);
        v8f d1 = __builtin_amdgcn_wmma_f32_16x16x4_f32(false, a1, false, b, (short)0, z, false, false);
        v8f d2 = __builtin_amdgcn_wmma_f32_16x16x4_f32(false, a2, false, b, (short)0, z, false, false);
        v8f d3 = __builtin_amdgcn_wmma_f32_16x16x4_f32(false, a3, false, b, (short)0, z, false, false);

        unsigned acc0 = inside_bits(d0, lo, hi);
        unsigned acc1 = inside_bits(d1, lo, hi);
        unsigned acc2 = inside_bits(d2, lo, hi);
        unsigned acc3 = inside_bits(d3, lo, hi);

        // lanes 0..3 gather rows k0+lane; 2 bits (l0+2u, l0+2u+1) per tile
        unsigned p0 = ((acc0 >> slot) & 1u) | (((acc0 >> (16 + slot)) & 1u) << 1);
        unsigned p1 = ((acc1 >> slot) & 1u) | (((acc1 >> (16 + slot)) & 1u) << 1);
        unsigned p2 = ((acc2 >> slot) & 1u) | (((acc2 >> (16 + slot)) & 1u) << 1);
        unsigned p3 = ((acc3 >> slot) & 1u) | (((acc3 >> (16 + slot)) & 1u) << 1);
        int sh = l0 & 31;                    // 0, 8, 16, 24
        rowword |= (p0 << sh) | (p1 << (sh + 2)) | (p2 << (sh + 4)) | (p3 << (sh + 6));

        if ((it & 3) == 3) {                 // 32 l-columns complete -> flush
            if (lane < 4) raw[(k0 + lane) * 32 + (l0 >> 5)] = rowword;
            rowword = 0;
        }

        a0 = n0; a1 = n1; a2 = n2; a3 = n3;
    }
}

// ---------------------------------------------------------------------------
// Kernel 3: fin[k][l] = ((raw&mask)|(raw&mask)^T) & tril, packed + row counts.
// ---------------------------------------------------------------------------
__global__ void finalize_kernel(const unsigned* __restrict__ raw,
                                const int* __restrict__ mask,
                                unsigned* __restrict__ fin,
                                int* __restrict__ rowcnt) {
    const int k = blockIdx.x;
    const int t = threadIdx.x;
    const int lane = t & 31;
    __shared__ int cnt;
    if (t == 0) cnt = 0;
    __syncthreads();

    int local = 0;
    for (int rep = 0; rep < 4; ++rep) {
        int l = rep * 256 + t;               // each wave covers 32 contiguous l
        bool bit = false;
        if (l <= k) {
            bool m_kl = ((raw[k * 32 + (l >> 5)] >> (l & 31)) & 1u) &&
                        (mask[k * KBOX + l] != 0);
            bool m_lk = ((raw[l * 32 + (k >> 5)] >> (k & 31)) & 1u) &&
                        (mask[l * KBOX + k] != 0);
            bit = m_kl || m_lk;
        }
        unsigned word = ballot32(bit);
        if (lane == 0) {
            fin[k * 32 + (l >> 5)] = word;
            local += __popc(word);
        }
    }
    if (lane == 0 && local) atomicAdd(&cnt, local);
    __syncthreads();
    if (t == 0) rowcnt[k] = cnt;
}

// ---------------------------------------------------------------------------
// Kernel 4: exclusive scan over the 1024 row counts (single block).
// ---------------------------------------------------------------------------
__global__ void scan_kernel(const int* __restrict__ rowcnt,
                            int* __restrict__ rowoff) {
    __shared__ int s[KBOX];
    int t = threadIdx.x;
    int v = rowcnt[t];
    s[t] = v;
    __syncthreads();
    for (int off = 1; off < KBOX; off <<= 1) {
        int add = (t >= off) ? s[t - off] : 0;
        __syncthreads();
        s[t] += add;
        __syncthreads();
    }
    rowoff[t] = s[t] - v;   // exclusive
}

// ---------------------------------------------------------------------------
// Kernel 5: pad the index region of d_out with -1.
// ---------------------------------------------------------------------------
__global__ void fill_kernel(float* __restrict__ out) {
    long long i = (long long)blockIdx.x * blockDim.x + threadIdx.x;
    long long total = (long long)KBOX * KBOX * 2;
    if (i < total) out[HPOSED_OUT + i] = -1.0f;
}

// ---------------------------------------------------------------------------
// Kernel 6: emit (l,k) pairs in jnp.nonzero row-major order.
// ---------------------------------------------------------------------------
__global__ void emit_kernel(const unsigned* __restrict__ fin,
                            const int* __restrict__ rowoff,
                            float* __restrict__ out) {
    const int k = blockIdx.x;
    const int t = threadIdx.x;   // 0..31
    __shared__ unsigned w[32];
    w[t] = fin[k * 32 + t];
    __syncthreads();
    int prefix = 0;
    for (int j = 0; j < t; ++j) prefix += __popc(w[j]);
    int pos = rowoff[k] + prefix;
    unsigned word = w[t];
    while (word) {
        int bbit = __ffs(word) - 1;
        int l = t * 32 + bbit;
        out[HPOSED_OUT + 2 * pos + 0] = (float)l;   // min = l (tril)
        out[HPOSED_OUT + 2 * pos + 1] = (float)k;   // max = k
        ++pos;
        word &= word - 1;
    }
}

// ---------------------------------------------------------------------------
extern "C" void kernel_launch(void* const* d_in, const int* in_sizes, int n_in,
                              void* d_out, int out_size, void* d_ws, size_t ws_size,
                              hipStream_t stream) {
    const float* bb_min = (const float*)d_in[0];
    const float* bb_max = (const float*)d_in[1];
    const float* xf     = (const float*)d_in[2];
    const int*   mask   = (const int*)d_in[3];
    float* out = (float*)d_out;

    char* ws = (char*)d_ws;
    float*    hposed = (float*)(ws + WS_HPOSED);
    float*    invT   = (float*)(ws + WS_INVT);
    unsigned* raw    = (unsigned*)(ws + WS_RAW);
    unsigned* fin    = (unsigned*)(ws + WS_FIN);
    int*      rowcnt = (int*)(ws + WS_CNT);
    int*      rowoff = (int*)(ws + WS_OFF);

    prep_kernel<<<KBOX / 256, 256, 0, stream>>>(bb_min, bb_max, xf,
                                                hposed, invT, out);
    pairs_wmma_kernel<<<KBOX / 4, 256, 0, stream>>>(hposed, invT,
                                                    bb_min, bb_max, raw);
    finalize_kernel<<<KBOX, 256, 0, stream>>>(raw, mask, fin, rowcnt);
    scan_kernel<<<1, KBOX, 0, stream>>>(rowcnt, rowoff);
    fill_kernel<<<(2 * KBOX * KBOX) / 256, 256, 0, stream>>>(out);
    emit_kernel<<<KBOX, 32, 0, stream>>>(fin, rowoff, out);
}